// VectorQuantStraightThrough_7679401525798
// MI455X (gfx1250) — compile-verified
//
#include <hip/hip_runtime.h>
#include <hip/hip_bf16.h>

typedef float v8f __attribute__((ext_vector_type(8)));
typedef float v2f __attribute__((ext_vector_type(2)));
typedef unsigned int u32x4 __attribute__((ext_vector_type(4)));
typedef int i32x8 __attribute__((ext_vector_type(8)));
typedef int i32x4 __attribute__((ext_vector_type(4)));

#define K_CODES 1024
#define D_DIM   256
#define HW      1024           // 32*32 spatial
#define CHW     (256 * 1024)   // per-batch stride in z_e
#define NPOS    8388608        // 32*256*32*32 elements per output tensor
#define IDX_OFF (2 * NPOS)
#define TILE_K  64             // codewords per LDS tile
#define ROWP    258            // padded LDS row stride (floats), bank-conflict free

// ---------------- kernel 0: codebook squared norms -> ws (4 KB) ----------------
__global__ __launch_bounds__(256) void vq_norms(const float* __restrict__ emb,
                                                float* __restrict__ norms) {
    int k = blockIdx.x * blockDim.x + threadIdx.x;
    if (k >= K_CODES) return;
    const float* row = emb + k * D_DIM;
    float s = 0.f;
#pragma unroll 8
    for (int c = 0; c < D_DIM; ++c) { float v = row[c]; s += v * v; }
    norms[k] = s;
}

// ---------------- kernel 1: TDM-staged fp32-WMMA score GEMM + argmin ----------------
// Block = 8 waves x 16 z-rows. Per 64-codeword tile: TDM DMAs emb tile into LDS
// (row-padded to 258 floats by the descriptor's pad feature), all waves run
// 64 k-steps x 4 chained v_wmma_f32_16x16x4_f32 with B from ds_load_b64.
__global__ __launch_bounds__(256) void vq_argmin(const float* __restrict__ z,
                                                 const float* __restrict__ emb,
                                                 const float* __restrict__ norms,
                                                 float* __restrict__ outIdx) {
    extern __shared__ float embS[];                         // [TILE_K][ROWP]

    const int lane     = threadIdx.x & 31;
    const int wave     = threadIdx.x >> 5;
    const int waveBase = (blockIdx.x * 8 + wave) * 16;      // first z-row of this wave
    const int b        = waveBase >> 10;                    // batch index (16 | waveBase)
    const int sp       = (waveBase & 1023) + (lane & 15);   // spatial pos of row M=lane&15
    const float* zbase = z + (size_t)b * CHW + sp;          // + ch*HW per channel
    const int chHalf   = (lane >> 4) << 1;                  // lanes 16-31 hold K=+2,+3

    float runMin[8];
    int   runIdx[8];
#pragma unroll
    for (int v = 0; v < 8; ++v) { runMin[v] = 3.0e38f; runIdx[v] = 0; }

    for (int nt = 0; nt < K_CODES / TILE_K; ++nt) {
#if __has_builtin(__builtin_amdgcn_tensor_load_to_lds)
        if (wave == 0) {
            // D# group0: count=1 | lds_addr | global_addr(57b) | type=2
            const unsigned long long gaddr =
                (unsigned long long)(const void*)(emb + nt * TILE_K * D_DIM);
            const unsigned lds_base = (unsigned)(__SIZE_TYPE__)(void*)embS;
            u32x4 g0;
            g0.x = 1u;                                        // count=1, user mode
            g0.y = lds_base;                                  // LDS byte address
            g0.z = (unsigned)(gaddr & 0xFFFFFFFFu);           // global_addr[31:0]
            g0.w = (unsigned)((gaddr >> 32) & 0x1FFFFFFu)     // global_addr[56:32]
                 | (2u << 30);                                // type=2 ("image")
            // D# group1: data_size=4B, pad every 256 dw by 2 dw -> row stride 258
            i32x8 g1;
            g1[0] = (2 << 16) | (1 << 20) | (7 << 22) | (1 << 25);
            g1[1] = (int)(256u << 16);        // tensor_dim0 = 256 (bits 79:48)
            g1[2] = (int)(1024u << 16);       // tensor_dim1 = 1024 (bits 111:80)
            g1[3] = (int)(256u << 16);        // tile_dim0 = 256 (bits 127:112)
            g1[4] = TILE_K;                   // tile_dim1 = 64, tile_dim2 = 0
            g1[5] = 256;                      // tensor_dim0_stride = 256
            g1[6] = (int)(256u << 16);        // tensor_dim1_stride = 256 (bits 223:208)
            g1[7] = 0;
            i32x4 gz4 = {0, 0, 0, 0};
            i32x8 gz8 = {0, 0, 0, 0, 0, 0, 0, 0};
            __builtin_amdgcn_tensor_load_to_lds(g0, g1, gz4, gz4, gz8, 0);
            __builtin_amdgcn_s_wait_tensorcnt(0);
        }
#else
        for (int i = threadIdx.x; i < TILE_K * D_DIM; i += 256) {
            int row = i >> 8, col = i & 255;
            embS[row * ROWP + col] = emb[(nt * TILE_K + row) * D_DIM + col];
        }
#endif
        __syncthreads();                                      // tile visible to all waves

        v8f acc[4];
#pragma unroll
        for (int sub = 0; sub < 4; ++sub) {
            float en = norms[nt * TILE_K + sub * 16 + (lane & 15)];
#pragma unroll
            for (int v = 0; v < 8; ++v) acc[sub][v] = en;     // C starts at |e_k|^2
        }
        for (int ks = 0; ks < 64; ++ks) {                     // K chunk of 4 per wmma
            const int ch = ks * 4 + chHalf;
            v2f a;                                            // A: row M=lane&15
            a.x = zbase[(size_t)ch * HW];
            a.y = zbase[(size_t)(ch + 1) * HW];
#pragma unroll
            for (int sub = 0; sub < 4; ++sub) {
                const v2f braw =
                    *(const v2f*)(embS + (sub * 16 + (lane & 15)) * ROWP + ch);
                v2f bm; bm.x = -2.0f * braw.x; bm.y = -2.0f * braw.y;
                acc[sub] = __builtin_amdgcn_wmma_f32_16x16x4_f32(
                    false, a, false, bm, (short)0, acc[sub], false, false);
            }
        }
#pragma unroll
        for (int sub = 0; sub < 4; ++sub) {
            const int kbase = nt * TILE_K + sub * 16 + (lane & 15);
#pragma unroll
            for (int v = 0; v < 8; ++v) {
                float d = acc[sub][v];
                if (d < runMin[v]) { runMin[v] = d; runIdx[v] = kbase; }
            }
        }
        __syncthreads();                                      // done reading tile
    }

    // reduce across the 16 lanes of each half (N dimension); lowest index wins ties
#pragma unroll
    for (int v = 0; v < 8; ++v) {
        float mv = runMin[v]; int mi = runIdx[v];
#pragma unroll
        for (int off = 1; off <= 8; off <<= 1) {
            float ov = __shfl_xor(mv, off, 32);
            int   oi = __shfl_xor(mi, off, 32);
            if (ov < mv || (ov == mv && oi < mi)) { mv = ov; mi = oi; }
        }
        if (lane == 0)  outIdx[waveBase + v]     = (float)mi;  // rows M=0..7
        if (lane == 16) outIdx[waveBase + 8 + v] = (float)mi;  // rows M=8..15
    }
}

// ---------------- kernel 2: gather codewords, write NCHW outputs ----------------
__global__ __launch_bounds__(256) void vq_gather(const float* __restrict__ emb,
                                                 float* __restrict__ out) {
    int pos = blockIdx.x * 256 + threadIdx.x;   // [b, c, w, h] flat, NPOS total
    int sp  = pos & 1023;
    int c   = (pos >> 10) & 255;
    int b   = pos >> 18;
    int n   = b * 1024 + sp;
    int k   = (int)out[IDX_OFF + n];
    float val = emb[k * D_DIM + c];
    out[pos]        = val;                      // z_q_st (== z_q in value)
    out[NPOS + pos] = val;                      // z_q
}

extern "C" void kernel_launch(void* const* d_in, const int* in_sizes, int n_in,
                              void* d_out, int out_size, void* d_ws, size_t ws_size,
                              hipStream_t stream) {
    const float* z   = (const float*)d_in[0];   // [32,256,32,32]
    const float* emb = (const float*)d_in[1];   // [1024,256]
    float* out   = (float*)d_out;
    float* norms = (float*)d_ws;                // 4 KB

    vq_norms<<<(K_CODES + 255) / 256, 256, 0, stream>>>(emb, norms);
    vq_argmin<<<256, 256, TILE_K * ROWP * sizeof(float), stream>>>(
        z, emb, norms, out + IDX_OFF);
    vq_gather<<<NPOS / 256, 256, 0, stream>>>(emb, out);
}